// VolumeRenderer_42296837931473
// MI455X (gfx1250) — compile-verified
//
#include <hip/hip_runtime.h>
#include <cstdint>

#define D_SAMPLES 128
#define C_CH 3
#define RAYS_PER_BLOCK 8
#define EPS_F 1e-10f
#define FAR_DELTA_F 1e10f

typedef float f32x4 __attribute__((ext_vector_type(4)));

// Low 32 bits of a flat pointer into __shared__ == LDS byte offset
// (flat LDS aperture keeps the offset in addr[31:0] per CDNA5 ISA 10.2).
__device__ __forceinline__ uint32_t lds_off(const void* p) {
    return (uint32_t)(uintptr_t)p;
}

// gfx1250 async global->LDS DMA copy, 16 bytes per lane. Tracked by ASYNCcnt.
// Non-temporal: the feature array is streamed exactly once.
__device__ __forceinline__ void async_copy_b128_nt(uint32_t lds, const float* g) {
    asm volatile("global_load_async_to_lds_b128 %0, %1, off th:TH_LOAD_NT"
                 :: "v"(lds), "v"((uint64_t)(uintptr_t)g)
                 : "memory");
}

__device__ __forceinline__ void wait_async0() {
    asm volatile("s_wait_asynccnt 0x0" ::: "memory");
}

__global__ __launch_bounds__(256) void VolumeRenderer_kernel(
    const float* __restrict__ density,      // (N, 128)
    const float* __restrict__ feature,      // (N, 128, 3)
    const float* __restrict__ depthv,       // (N, 128)
    float* __restrict__ out_feat,           // (N, 3)
    float* __restrict__ out_depth,          // (N,)
    int N)
{
    __shared__ __align__(16) float s_feat[RAYS_PER_BLOCK * D_SAMPLES * C_CH];

    const int tid  = threadIdx.x;
    const int wave = tid >> 5;          // one wave32 per ray
    const int lane = tid & 31;          // lane owns samples 4*lane .. 4*lane+3
    const int ray  = blockIdx.x * RAYS_PER_BLOCK + wave;
    if (ray >= N) return;

    const size_t base_nd  = (size_t)ray * D_SAMPLES;
    const size_t base_ndc = (size_t)ray * (D_SAMPLES * C_CH);

    // ---- Kick off the big (N,128,3) operand via the gfx1250 async
    //      global->LDS DMA path; it is not needed until after the scan. ----
    float* lf = &s_feat[wave * (D_SAMPLES * C_CH) + lane * 12];
    async_copy_b128_nt(lds_off(lf + 0), feature + base_ndc + lane * 12 + 0);
    async_copy_b128_nt(lds_off(lf + 4), feature + base_ndc + lane * 12 + 4);
    async_copy_b128_nt(lds_off(lf + 8), feature + base_ndc + lane * 12 + 8);

    // ---- Density/depth straight to VGPRs (coalesced b128, non-temporal) ----
    const f32x4 dn = __builtin_nontemporal_load(
        (const f32x4*)(density + base_nd + lane * 4));  // densities s0..s3
    const f32x4 dp = __builtin_nontemporal_load(
        (const f32x4*)(depthv  + base_nd + lane * 4));  // depths    s0..s3

    // ---- deltas (last sample of ray gets FAR_DELTA) ----
    const float depn = __shfl_down(dp.x, 1, 32);  // depth[4*(lane+1)]
    const float dl0 = dp.y - dp.x;
    const float dl1 = dp.z - dp.y;
    const float dl2 = dp.w - dp.z;
    const float dl3 = (lane == 31) ? FAR_DELTA_F : (depn - dp.w);

    // ---- s = -sigma*delta + eps ----
    const float sv0 = fmaf(-dn.x, dl0, EPS_F);
    const float sv1 = fmaf(-dn.y, dl1, EPS_F);
    const float sv2 = fmaf(-dn.z, dl2, EPS_F);
    const float sv3 = fmaf(-dn.w, dl3, EPS_F);

    // ---- exclusive cumsum along D: serial-4 prefix + wave32 scan ----
    const float e1 = sv0;
    const float e2 = sv0 + sv1;
    const float e3 = e2 + sv2;
    const float tot = e3 + sv3;
    float incl = tot;
    #pragma unroll
    for (int o = 1; o < 32; o <<= 1) {
        float y = __shfl_up(incl, o, 32);
        if (lane >= o) incl += y;
    }
    const float base = incl - tot;  // exclusive prefix of lane totals

    // ---- transmittance & weights: T_{i+1} = T_i * exp(s_i); w_i = T_i - T_{i+1} ----
    const float es0 = __expf(sv0);
    const float es1 = __expf(sv1);
    const float es2 = __expf(sv2);
    const float es3 = __expf(sv3);
    const float T0 = __expf(base);
    const float T1 = T0 * es0;
    const float T2 = T1 * es1;
    const float T3 = T2 * es2;
    const float T4 = T3 * es3;
    const float w0 = T0 - T1;
    const float w1 = T1 - T2;
    const float w2 = T2 - T3;
    const float w3 = T3 - T4;

    // depth output needs no feature data; fold it in while DMA completes
    float pd = w0 * dp.x + w1 * dp.y + w2 * dp.z + w3 * dp.w;

    // ---- now consume the async-staged feature tile ----
    wait_async0();   // wave only reads LDS it wrote itself -> no barrier needed
    const f32x4 f0 = *(const f32x4*)(lf + 0); // s0c0 s0c1 s0c2 s1c0
    const f32x4 f1 = *(const f32x4*)(lf + 4); // s1c1 s1c2 s2c0 s2c1
    const f32x4 f2 = *(const f32x4*)(lf + 8); // s2c2 s3c0 s3c1 s3c2

    float pf0 = w0 * f0.x + w1 * f0.w + w2 * f1.z + w3 * f2.y;
    float pf1 = w0 * f0.y + w1 * f1.x + w2 * f1.w + w3 * f2.z;
    float pf2 = w0 * f0.z + w1 * f1.y + w2 * f2.x + w3 * f2.w;

    // ---- wave32 butterfly reduction ----
    #pragma unroll
    for (int o = 16; o > 0; o >>= 1) {
        pf0 += __shfl_xor(pf0, o, 32);
        pf1 += __shfl_xor(pf1, o, 32);
        pf2 += __shfl_xor(pf2, o, 32);
        pd  += __shfl_xor(pd,  o, 32);
    }

    if (lane == 0) {
        out_feat[(size_t)ray * 3 + 0] = pf0;
        out_feat[(size_t)ray * 3 + 1] = pf1;
        out_feat[(size_t)ray * 3 + 2] = pf2;
        out_depth[ray] = pd;
    }
}

extern "C" void kernel_launch(void* const* d_in, const int* in_sizes, int n_in,
                              void* d_out, int out_size, void* d_ws, size_t ws_size,
                              hipStream_t stream) {
    const float* density = (const float*)d_in[0];  // (N, 128)
    const float* feature = (const float*)d_in[1];  // (N, 128, 3)
    const float* depthv  = (const float*)d_in[2];  // (N, 128)

    const int N = in_sizes[0] / D_SAMPLES;

    float* out_feat  = (float*)d_out;              // first N*3 floats
    float* out_depth = (float*)d_out + (size_t)N * 3;

    const int blocks = (N + RAYS_PER_BLOCK - 1) / RAYS_PER_BLOCK;
    VolumeRenderer_kernel<<<blocks, 256, 0, stream>>>(
        density, feature, depthv, out_feat, out_depth, N);
}